// CSDNet_55594056679461
// MI455X (gfx1250) — compile-verified
//
#include <hip/hip_runtime.h>
#include <hip/hip_bf16.h>

typedef __attribute__((ext_vector_type(2))) float v2f;
typedef __attribute__((ext_vector_type(8))) float v8f;

__device__ __forceinline__ v8f wmma4(v2f a, v2f b, v8f c) {
  // D = A(16x4) * B(4x16) + C, full f32
  return __builtin_amdgcn_wmma_f32_16x16x4_f32(false, a, false, b, (short)0, c, false, false);
}

// ---------------- workspace layout (float offsets) ----------------
#define NVOX48   110592           // 48^3
#define OFF_AINV47T 0             // 48*48
#define OFF_AINV18  2304          // 18*18 (=324)
#define OFF_AQTB    4096          // 110592*48
#define OFF_CA      5312512       // 110592*48
#define OFF_CB      10620928      // 110592*48
#define OFF_X1      15929344     // 97336*96

// K-paired B staging: Bp[((k>>1)*N + n)*2 + (k&1)] = B[k][n]
// -> one aligned ds_load_b64 yields the (K,K+1) fragment pair per lane.

// ================= prep: build + invert A47 (padded 48) and A18 =================
__global__ __launch_bounds__(64) void prep_kernel(const float* __restrict__ AQ,
                                                  const float* __restrict__ dreg,
                                                  float* __restrict__ ainv47T,
                                                  float* __restrict__ ainv18) {
  __shared__ float M[48 * 48];
  __shared__ float Iv[48 * 48];
  int t = threadIdx.x;

  if (t < 48) {
    for (int c = 0; c < 48; ++c) {
      float s;
      if (t < 47 && c < 47) {
        s = 0.f;
        for (int n = 0; n < 300; ++n) s += AQ[n * 47 + t] * AQ[n * 47 + c];
        if (c == t) s += dreg[0];
      } else {
        s = (c == t) ? 1.f : 0.f;
      }
      M[t * 48 + c] = s;
      Iv[t * 48 + c] = (c == t) ? 1.f : 0.f;
    }
  }
  __syncthreads();
  for (int p = 0; p < 48; ++p) {
    float pinv = 1.f / M[p * 48 + p];
    __syncthreads();
    if (t < 48 && t != p) {
      float f = M[t * 48 + p] * pinv;
      for (int c = 0; c < 48; ++c) {
        M[t * 48 + c] -= f * M[p * 48 + c];
        Iv[t * 48 + c] -= f * Iv[p * 48 + c];
      }
    }
    __syncthreads();
    if (t == p) {
      for (int c = 0; c < 48; ++c) { M[p * 48 + c] *= pinv; Iv[p * 48 + c] *= pinv; }
    }
    __syncthreads();
  }
  // store transposed, K-PAIRED: ainv47T[((k>>1)*48+n)*2+(k&1)] = Ainv[n][k]
  if (t < 48) {
    for (int n = 0; n < 48; ++n) {
      float v = (t < 47 && n < 47) ? Iv[n * 48 + t] : 0.f;
      ainv47T[(((t >> 1) * 48) + n) * 2 + (t & 1)] = v;
    }
  }
  __syncthreads();

  // ---- A18 over IDX = [0..15, 45, 46] ----
  if (t < 18) {
    int rr = (t < 16) ? t : 29 + t;
    for (int c = 0; c < 18; ++c) {
      int cc = (c < 16) ? c : 29 + c;
      float s = 0.f;
      for (int n = 0; n < 300; ++n) s += AQ[n * 47 + rr] * AQ[n * 47 + cc];
      if (c == t) s += 0.01f;
      M[t * 48 + c] = s;
      Iv[t * 48 + c] = (c == t) ? 1.f : 0.f;
    }
  }
  __syncthreads();
  for (int p = 0; p < 18; ++p) {
    float pinv = 1.f / M[p * 48 + p];
    __syncthreads();
    if (t < 18 && t != p) {
      float f = M[t * 48 + p] * pinv;
      for (int c = 0; c < 18; ++c) {
        M[t * 48 + c] -= f * M[p * 48 + c];
        Iv[t * 48 + c] -= f * Iv[p * 48 + c];
      }
    }
    __syncthreads();
    if (t == p) {
      for (int c = 0; c < 18; ++c) { M[p * 48 + c] *= pinv; Iv[p * 48 + c] *= pinv; }
    }
    __syncthreads();
  }
  if (t < 18)
    for (int c = 0; c < 18; ++c) ainv18[t * 18 + c] = Iv[t * 48 + c];
}

// ================= AQ_Tb: [110592 x 300] x [300 x 47] via WMMA f32 =================
__global__ __launch_bounds__(256) void aqtb_kernel(const float* __restrict__ b,
                                                   const float* __restrict__ AQ,
                                                   float* __restrict__ out) {
  __shared__ __align__(16) float Bs[60 * 48];  // K-paired chunk of AQ (60 k x 48 n)
  int t = threadIdx.x, lane = t & 31, wv = t >> 5;
  int hh = lane >> 4, ln = lane & 15;
  int m0 = (blockIdx.x * 8 + wv) * 16;
  const float* arow = b + (size_t)(m0 + ln) * 300;

  v8f acc[3];
  v8f z = {0.f, 0.f, 0.f, 0.f, 0.f, 0.f, 0.f, 0.f};
  acc[0] = z; acc[1] = z; acc[2] = z;

  for (int ch = 0; ch < 5; ++ch) {
    int kbase = ch * 60;
    for (int j = t; j < 60 * 48; j += 256) {
      int k = j / 48, n = j % 48;
      Bs[(((k >> 1) * 48) + n) * 2 + (k & 1)] =
          (n < 47) ? AQ[(kbase + k) * 47 + n] : 0.f;
    }
    __syncthreads();
    __builtin_prefetch(arow + kbase + 64, 0, 0);  // global_prefetch next chunk
#pragma unroll 5
    for (int s = 0; s < 15; ++s) {
      int kb = s * 4 + 2 * hh;
      v2f a = *(const v2f*)(arow + kbase + kb);
#pragma unroll
      for (int nt = 0; nt < 3; ++nt) {
        v2f bb = *(const v2f*)(Bs + (((kb >> 1) * 48) + nt * 16 + ln) * 2);
        acc[nt] = wmma4(a, bb, acc[nt]);
      }
    }
    __syncthreads();
  }
  float* orow = out + (size_t)m0 * 48;
#pragma unroll
  for (int r = 0; r < 8; ++r) {
    int m = r + 8 * hh;
#pragma unroll
    for (int nt = 0; nt < 3; ++nt)
      orow[(size_t)m * 48 + nt * 16 + ln] = acc[nt][r];
  }
}

// ================= c init: c = scatter(Ainv18 * AQ_Tb[IDX]) =================
__global__ __launch_bounds__(256) void cinit_kernel(const float* __restrict__ aqtb,
                                                    const float* __restrict__ ainv18,
                                                    float* __restrict__ c) {
  __shared__ float Ai[18 * 18];
  int t = threadIdx.x;
  for (int j = t; j < 18 * 18; j += 256) Ai[j] = ainv18[j];
  __syncthreads();
  int v = blockIdx.x * 256 + t;  // 432*256 == 110592 exactly
  const float* row = aqtb + (size_t)v * 48;
  float r[18];
#pragma unroll
  for (int j = 0; j < 16; ++j) r[j] = row[j];
  r[16] = row[45]; r[17] = row[46];
  float o[18];
#pragma unroll
  for (int i = 0; i < 18; ++i) {
    float s = 0.f;
#pragma unroll
    for (int j = 0; j < 18; ++j) s += Ai[i * 18 + j] * r[j];
    o[i] = s;
  }
  float* cr = c + (size_t)v * 48;
#pragma unroll
  for (int k = 0; k < 48; ++k) cr[k] = 0.f;
#pragma unroll
  for (int i = 0; i < 16; ++i) cr[i] = o[i];
  cr[45] = o[16]; cr[46] = o[17];
}

// ================= conv1: 3x3x3 VALID, 47->96 ch, ReLU, via WMMA =================
__global__ __launch_bounds__(256) void conv1_kernel(const float* __restrict__ cin,
                                                    const float* __restrict__ w1,
                                                    const float* __restrict__ b1,
                                                    float* __restrict__ x1,
                                                    int Dn, int Do) {
  __shared__ __align__(16) float Ws[2][48 * 96];  // double-buffered tap, K-paired
  int t = threadIdx.x, lane = t & 31, wv = t >> 5;
  int hh = lane >> 4, ln = lane & 15;
  const int ntx = 3;  // Do <= 46 -> 3 x-tiles of 16
  int gtile = blockIdx.x * 8 + wv;
  int ntiles = Do * Do * ntx;
  bool valid = gtile < ntiles;
  int vt = valid ? gtile : 0;
  int tx = vt % ntx, rowid = vt / ntx;
  int oy = rowid % Do, oz = rowid / Do;
  int ox0 = tx * 16;
  int oxc = ox0 + ln; if (oxc >= Do) oxc = Do - 1;

  auto stage = [&](int tap) {
    float* W = Ws[tap & 1];
    for (int j = t; j < 48 * 96; j += 256) {
      int k = j / 96, n = j % 96;
      W[(((k >> 1) * 96) + n) * 2 + (k & 1)] =
          (k < 47) ? w1[(n * 47 + k) * 27 + tap] : 0.f;
    }
  };

  v8f acc[6];
  v8f z = {0.f, 0.f, 0.f, 0.f, 0.f, 0.f, 0.f, 0.f};
#pragma unroll
  for (int i = 0; i < 6; ++i) acc[i] = z;

  stage(0);
  for (int tap = 0; tap < 27; ++tap) {
    __syncthreads();                 // Ws[tap&1] ready; prior compute done
    if (tap + 1 < 27) stage(tap + 1);
    if (valid) {
      int dz = tap / 9, dy = (tap / 3) % 3, dx = tap % 3;
      const float* W = Ws[tap & 1];
      const float* abase =
          cin + (size_t)(((oz + dz) * Dn + (oy + dy)) * Dn + (oxc + dx)) * 48;
#pragma unroll 4
      for (int s = 0; s < 12; ++s) {
        int kb = s * 4 + 2 * hh;
        v2f a = *(const v2f*)(abase + kb);
#pragma unroll
        for (int nt = 0; nt < 6; ++nt) {
          v2f bb = *(const v2f*)(W + (((kb >> 1) * 96) + nt * 16 + ln) * 2);
          acc[nt] = wmma4(a, bb, acc[nt]);
        }
      }
    }
  }
  if (valid) {
    float bias[6];
#pragma unroll
    for (int nt = 0; nt < 6; ++nt) bias[nt] = b1[nt * 16 + ln];
    float* obase = x1 + (size_t)((oz * Do + oy) * Do) * 96;
#pragma unroll
    for (int r = 0; r < 8; ++r) {
      int m = r + 8 * hh, ox = ox0 + m;
      if (ox < Do) {
#pragma unroll
        for (int nt = 0; nt < 6; ++nt) {
          float v = acc[nt][r] + bias[nt];
          obase[(size_t)ox * 96 + nt * 16 + ln] = v > 0.f ? v : 0.f;
        }
      }
    }
  }
}

// ======== conv2(relu) + conv3 + gate + rhs + solve, fused per 16-voxel tile ========
__global__ __launch_bounds__(128) void conv23_kernel(const float* __restrict__ x1,
                                                     const float* __restrict__ aqtb,
                                                     const float* __restrict__ cprev,
                                                     const float* __restrict__ w2,
                                                     const float* __restrict__ b2,
                                                     const float* __restrict__ w3,
                                                     const float* __restrict__ b3,
                                                     const float* __restrict__ ainvT,
                                                     float* __restrict__ cout,
                                                     const float* __restrict__ dregp,
                                                     int Do, int casc) {
  __shared__ __align__(16) float R1[96 * 96];      // W2p / W3p / rhs+AinvTp
  __shared__ __align__(16) float R2[4 * 16 * 98];  // per-wave 16x96 staging, stride 98
  __shared__ float BS[192];                        // biases b2 | b3(pad)
  int t = threadIdx.x, lane = t & 31, wv = t >> 5;
  int hh = lane >> 4, ln = lane & 15;
  int nvox = Do * Do * Do;
  int ntiles = (nvox + 15) >> 4;
  int gtile = blockIdx.x * 4 + wv;
  bool valid = gtile < ntiles;
  int v0 = (valid ? gtile : 0) * 16;
  float dreg = dregp[0];
  float* stg = R2 + wv * 16 * 98;
  v8f z = {0.f, 0.f, 0.f, 0.f, 0.f, 0.f, 0.f, 0.f};

  // stage W2^T K-paired
  for (int j = t; j < 96 * 96; j += 128) {
    int k = j / 96, n = j % 96;
    R1[(((k >> 1) * 96) + n) * 2 + (k & 1)] = w2[n * 96 + k];
  }
  for (int j = t; j < 192; j += 128) BS[j] = (j < 96) ? b2[j] : ((j - 96) < 94 ? b3[j - 96] : 0.f);
  __syncthreads();

  // ---- GEMM1: x2 = relu(x1 * W2 + b2) ----
  {
    v8f acc[6];
#pragma unroll
    for (int i = 0; i < 6; ++i) acc[i] = z;
    int m = v0 + ln; if (m >= nvox) m = nvox - 1;
    const float* arow = x1 + (size_t)m * 96;
#pragma unroll 4
    for (int s = 0; s < 24; ++s) {
      int kb = s * 4 + 2 * hh;
      v2f a = *(const v2f*)(arow + kb);
#pragma unroll
      for (int nt = 0; nt < 6; ++nt) {
        v2f bb = *(const v2f*)(R1 + (((kb >> 1) * 96) + nt * 16 + ln) * 2);
        acc[nt] = wmma4(a, bb, acc[nt]);
      }
    }
#pragma unroll
    for (int r = 0; r < 8; ++r) {
      int mm = r + 8 * hh;
#pragma unroll
      for (int nt = 0; nt < 6; ++nt) {
        int n = nt * 16 + ln;
        float v = acc[nt][r] + BS[n];
        stg[mm * 98 + n] = v > 0.f ? v : 0.f;
      }
    }
  }
  __syncthreads();
  // stage W3^T K-paired (cols >= 94 zero)
  for (int j = t; j < 96 * 96; j += 128) {
    int k = j / 96, n = j % 96;
    R1[(((k >> 1) * 96) + n) * 2 + (k & 1)] = (n < 94) ? w3[n * 96 + k] : 0.f;
  }
  __syncthreads();

  // ---- GEMM2: cc = x2 * W3 + b3 (back into stg) ----
  {
    v8f acc[6];
#pragma unroll
    for (int i = 0; i < 6; ++i) acc[i] = z;
#pragma unroll 4
    for (int s = 0; s < 24; ++s) {
      int kb = s * 4 + 2 * hh;
      v2f a = *(const v2f*)(stg + ln * 98 + kb);
#pragma unroll
      for (int nt = 0; nt < 6; ++nt) {
        v2f bb = *(const v2f*)(R1 + (((kb >> 1) * 96) + nt * 16 + ln) * 2);
        acc[nt] = wmma4(a, bb, acc[nt]);
      }
    }
#pragma unroll
    for (int r = 0; r < 8; ++r) {
      int mm = r + 8 * hh;
#pragma unroll
      for (int nt = 0; nt < 6; ++nt) {
        int n = nt * 16 + ln;
        stg[mm * 98 + n] = acc[nt][r] + BS[96 + n];
      }
    }
  }
  __syncthreads();

  // ---- gate + rhs (into R1 low), stage K-paired AinvT (into R1+4096) ----
  {
    float* rhs = R1 + wv * 768;  // 16 x 48
    int coff = casc + 1, Dn = Do + 2;
#pragma unroll
    for (int e = 0; e < 24; ++e) {
      int idx = lane + 32 * e;
      int mm = idx / 48, k = idx % 48;
      int v = v0 + mm; if (v >= nvox) v = nvox - 1;
      int oz = v / (Do * Do), rem = v % (Do * Do), oy = rem / Do, ox = rem % Do;
      float val = 0.f;
      if (k < 47) {
        float a0 = stg[mm * 98 + k];
        float a1 = stg[mm * 98 + 47 + k];
        float g = a0 * (1.f / (1.f + __expf(-a1)));
        if (casc > 0)
          g += cprev[(size_t)(((oz + 1) * Dn + (oy + 1)) * Dn + (ox + 1)) * 48 + k];
        float r0 = aqtb[(size_t)(((oz + coff) * 48 + (oy + coff)) * 48 + (ox + coff)) * 48 + k];
        val = r0 + dreg * g;
      }
      rhs[mm * 48 + k] = val;
    }
    float* AiT = R1 + 4096;
    for (int j = t; j < 48 * 48; j += 128) AiT[j] = ainvT[j];  // already K-paired
  }
  __syncthreads();

  // ---- GEMM3: c_new = rhs * Ainv^T ----
  {
    float* rhs = R1 + wv * 768;
    float* AiT = R1 + 4096;
    v8f acc[3];
    acc[0] = z; acc[1] = z; acc[2] = z;
#pragma unroll
    for (int s = 0; s < 12; ++s) {
      int kb = s * 4 + 2 * hh;
      v2f a = *(const v2f*)(rhs + ln * 48 + kb);
#pragma unroll
      for (int nt = 0; nt < 3; ++nt) {
        v2f bb = *(const v2f*)(AiT + (((kb >> 1) * 48) + nt * 16 + ln) * 2);
        acc[nt] = wmma4(a, bb, acc[nt]);
      }
    }
#pragma unroll
    for (int r = 0; r < 8; ++r) {
      int mm = r + 8 * hh, v = v0 + mm;
      if (valid && v < nvox) {
#pragma unroll
        for (int nt = 0; nt < 3; ++nt)
          cout[(size_t)v * 48 + nt * 16 + ln] = acc[nt][r];
      }
    }
  }
}

// ================= strip channel padding to d_out (40^3 x 47) =================
__global__ __launch_bounds__(256) void final_copy_kernel(const float* __restrict__ c,
                                                         float* __restrict__ out, int n) {
  int i = blockIdx.x * 256 + threadIdx.x;
  if (i < n) {
    int v = i / 47, k = i % 47;
    out[i] = c[(size_t)v * 48 + k];
  }
}

extern "C" void kernel_launch(void* const* d_in, const int* in_sizes, int n_in,
                              void* d_out, int out_size, void* d_ws, size_t ws_size,
                              hipStream_t stream) {
  const float* b    = (const float*)d_in[0];
  const float* AQ   = (const float*)d_in[1];
  const float* w1   = (const float*)d_in[2];
  const float* b1   = (const float*)d_in[3];
  const float* w2   = (const float*)d_in[4];
  const float* b2   = (const float*)d_in[5];
  const float* w3   = (const float*)d_in[6];
  const float* b3   = (const float*)d_in[7];
  const float* dreg = (const float*)d_in[8];

  float* ws      = (float*)d_ws;
  float* ainv47T = ws + OFF_AINV47T;
  float* ainv18  = ws + OFF_AINV18;
  float* aqtb    = ws + OFF_AQTB;
  float* cA      = ws + OFF_CA;
  float* cB      = ws + OFF_CB;
  float* x1      = ws + OFF_X1;

  prep_kernel<<<1, 64, 0, stream>>>(AQ, dreg, ainv47T, ainv18);
  aqtb_kernel<<<NVOX48 / (8 * 16), 256, 0, stream>>>(b, AQ, aqtb);
  cinit_kernel<<<NVOX48 / 256, 256, 0, stream>>>(aqtb, ainv18, cA);

  float* cin = cA;
  float* cou = cB;
  for (int i = 0; i < 4; ++i) {
    int Dn = 48 - 2 * i, Do = Dn - 2;
    int tiles1 = Do * Do * 3;
    conv1_kernel<<<(tiles1 + 7) / 8, 256, 0, stream>>>(
        cin, w1 + (size_t)i * 96 * 47 * 27, b1 + (size_t)i * 96, x1, Dn, Do);
    int nvox = Do * Do * Do;
    int tiles2 = (nvox + 15) / 16;
    conv23_kernel<<<(tiles2 + 3) / 4, 128, 0, stream>>>(
        x1, aqtb, cin, w2 + (size_t)i * 96 * 96, b2 + (size_t)i * 96,
        w3 + (size_t)i * 94 * 96, b3 + (size_t)i * 94, ainv47T, cou, dreg, Do, i);
    float* tmp = cin; cin = cou; cou = tmp;
  }
  int n = 40 * 40 * 40 * 47;
  final_copy_kernel<<<(n + 255) / 256, 256, 0, stream>>>(cin, (float*)d_out, n);
}